// FHNet_74500502717129
// MI455X (gfx1250) — compile-verified
//
#include <hip/hip_runtime.h>

// FitzHugh-Nagumo network for MI455X (gfx1250, wave32).
//  k0: pack W1 -> zero-padded WMMA-fragment order  W1t[k/4][112][4]
//  k1: z1 = batch @ W1^T via V_WMMA_F32_16X16X4_F32, 2 M-tiles x 7 N-tiles/wave
//  k2: fused scan1 -> GEMM2 -> scan2, async global->LDS double-buffered staging

typedef __attribute__((ext_vector_type(2))) float v2f;
typedef __attribute__((ext_vector_type(8))) float v8f;

#define B_SZ  64
#define T_SZ  2048
#define K1    784
#define N1    100
#define N1P   112        // N padded to 7 tiles of 16
#define N2    10
#define NT1   7
#define FHN_DT 0.04f
#define CH    32         // time-steps per chunk in kernel 2
#define NCHUNK (T_SZ / CH)
#define ZWORDS (CH * N1) // 3200 floats per staged chunk

// ---------------------------------------------------------------------------
// Kernel 0: pack W1 (100x784) into padded fragment order:
//   W1t[(k>>2)*112 + n][k&3],  n in [0,112), zero for n >= 100.
// A wave's 7 B-fragments for one K-group then live in one 1792 B block.
// ---------------------------------------------------------------------------
__global__ __launch_bounds__(256) void fhn_pack_w1(
    const float* __restrict__ W1, float* __restrict__ W1t)
{
  int idx = blockIdx.x * 256 + threadIdx.x;      // over 112*784
  if (idx >= N1P * K1) return;
  int n = idx / K1;
  int k = idx - n * K1;
  float v = (n < N1) ? W1[n * K1 + k] : 0.0f;
  W1t[((k >> 2) * N1P + n) * 4 + (k & 3)] = v;
}

// ---------------------------------------------------------------------------
// Kernel 1: GEMM1.  M = 131072, K = 784, N = 112(padded).
// Each wave: 2 M-tiles x all 7 N-tiles -> 14 WMMAs per 9 loads per K-step;
// every element of batch (411 MB) read exactly once; W1t (351 KB) lives in L2.
// V_WMMA_F32_16X16X4_F32 layout (ISA 7.12.2):
//   A 16x4 : lanes 0-15 {K0,K1} of row M=lane ; lanes 16-31 {K2,K3}
//   B 4x16 : lanes 0-15 col N=lane {K0,K1}    ; lanes 16-31 {K2,K3}
//   C/D    : VGPR r -> (M=r, N=lane) lanes 0-15 ; (M=r+8) lanes 16-31
// ---------------------------------------------------------------------------
__global__ __launch_bounds__(256) void fhn_gemm1_wmma(
    const float* __restrict__ A,     // (131072, 784)
    const float* __restrict__ W1t,   // packed (196, 112, 4)
    float* __restrict__ z1)          // (131072, 100)
{
  const int lane  = threadIdx.x & 31;
  const int wave  = threadIdx.x >> 5;
  const int w     = blockIdx.x * 8 + wave;     // 512 blocks * 8 waves = 4096
  const int mrow  = lane & 15;
  const int khalf = (lane >> 4) << 1;          // 0 / 2

  const float* arow0 = A + (w * 32 + mrow) * K1 + khalf;
  const float* arow1 = arow0 + 16 * K1;
  const float* bbase = W1t + mrow * 4 + khalf; // + kg*448 + j*64 per fragment

  v8f acc[2][NT1] = {};

  for (int kg = 0; kg < K1 / 4; ++kg) {
    __builtin_prefetch(arow0 + kg * 4 + 128, 0, 3);  // stream A ahead (near scope)
    __builtin_prefetch(arow1 + kg * 4 + 128, 0, 3);
    v2f a0 = *(const v2f*)(arow0 + kg * 4);
    v2f a1 = *(const v2f*)(arow1 + kg * 4);
    const float* bk = bbase + kg * (N1P * 4);
#pragma unroll
    for (int j = 0; j < NT1; ++j) {
      v2f bf = *(const v2f*)(bk + j * 64);
      acc[0][j] = __builtin_amdgcn_wmma_f32_16x16x4_f32(
          false, a0, false, bf, (short)0, acc[0][j], false, false);
      acc[1][j] = __builtin_amdgcn_wmma_f32_16x16x4_f32(
          false, a1, false, bf, (short)0, acc[1][j], false, false);
    }
  }

  const int mb = w * 32 + ((lane >> 4) << 3);  // +8 for upper half-wave
#pragma unroll
  for (int t = 0; t < 2; ++t) {
#pragma unroll
    for (int j = 0; j < NT1; ++j) {
      int n = j * 16 + mrow;
      if (n < N1) {
#pragma unroll
        for (int r = 0; r < 8; ++r)
          z1[(mb + t * 16 + r) * N1 + n] = acc[t][j][r];
      }
    }
  }
}

// ---------------------------------------------------------------------------
// Kernel 2: fused scan1(100ch) -> GEMM2(100->10) -> scan2(10ch).
// One block per batch element; z1 chunks stream into LDS with async
// global->LDS loads, double-buffered so chunk c+1 is in flight while the
// sequential FHN chain processes chunk c.
// ---------------------------------------------------------------------------
__global__ __launch_bounds__(128) void fhn_scan_fused(
    const float* __restrict__ z1,   // (B, T, 100)
    const float* __restrict__ W2,   // (10, 100)
    float* __restrict__ out)        // (B, T, 10)
{
  __shared__ float zstage[2][ZWORDS];   // async-staged z1 chunks
  __shared__ float Vbuf[CH][N1 + 1];    // 0.5*V1 chunk, stride 101 -> no conflicts
  __shared__ float W2s[N2 * N1];
  __shared__ float z2buf[CH][N2];

  const int tid = threadIdx.x;
  const int b   = blockIdx.x;
  const float* zbase = z1 + b * T_SZ * N1;

  for (int i = tid; i < N2 * N1; i += 128) W2s[i] = W2[i];

  // prologue: stage chunk 0 (25 async loads per thread/wave)
#pragma unroll
  for (int p = 0; p < ZWORDS / 128; ++p) {
    int i = tid + p * 128;
    unsigned lds_off = (unsigned)(size_t)&zstage[0][i];
    unsigned long long ga = (unsigned long long)(size_t)(zbase + i);
    asm volatile("global_load_async_to_lds_b32 %0, %1, off"
                 :: "v"(lds_off), "v"(ga) : "memory");
  }

  const float one_dt = 1.0f + FHN_DT;
  const float dt3    = FHN_DT / 3.0f;

  float V1 = 0.0f, S1 = 0.0f;   // scan1 state (tid < 100)
  float V2 = 0.0f, S2 = 0.0f;   // scan2 state (tid < 10)

  for (int c = 0; c < NCHUNK; ++c) {
    const int buf = c & 1;

    if (c + 1 < NCHUNK) {
      // kick off next chunk, then wait until only those 25 remain outstanding
      const float* src = zbase + (c + 1) * ZWORDS;
#pragma unroll
      for (int p = 0; p < ZWORDS / 128; ++p) {
        int i = tid + p * 128;
        unsigned lds_off = (unsigned)(size_t)&zstage[buf ^ 1][i];
        unsigned long long ga = (unsigned long long)(size_t)(src + i);
        asm volatile("global_load_async_to_lds_b32 %0, %1, off"
                     :: "v"(lds_off), "v"(ga) : "memory");
      }
      asm volatile("s_wait_asynccnt 0x19" ::: "memory");   // <=25 : chunk c done
    } else {
      asm volatile("s_wait_asynccnt 0x0" ::: "memory");    // last chunk done
    }
    __syncthreads();

    // ---- phase A: 32 FHN steps, layer-1 channel n = tid, z from LDS --------
    if (tid < N1) {
#pragma unroll
      for (int k = 0; k < CH; ++k) {
        float z = zstage[buf][k * N1 + tid];
        Vbuf[k][tid] = 0.5f * V1;                       // emit V1[t], pre-scaled
        float vv   = V1 * V1;
        float coef = fmaf(-dt3, vv, one_dt);            // (1+dt) - dt/3 V^2
        float Vn   = fmaf(V1, coef, FHN_DT * (z - S1));
        float Sn   = fmaf(FHN_DT * 0.08f, fmaf(-0.8f, S1, V1 + 0.7f), S1);
        V1 = Vn; S1 = Sn;
      }
    }
    __syncthreads();

    // ---- phase B: z2[k][m] = dot(0.5*V1[t0+k], W2[m])  (320 dots) ----------
#pragma unroll
    for (int p = 0; p < 3; ++p) {
      int idx = tid + p * 128;
      if (idx < CH * N2) {
        int k = idx & 31, m = idx >> 5;
        float s = 0.0f;
#pragma unroll 5
        for (int n = 0; n < N1; ++n) s = fmaf(Vbuf[k][n], W2s[m * N1 + n], s);
        z2buf[k][m] = s;
      }
    }
    __syncthreads();

    // ---- phase C: 32 FHN steps, layer-2 channel m = tid, stream V2 ---------
    if (tid < N2) {
      float* op = out + (b * T_SZ + c * CH) * N2 + tid;
#pragma unroll
      for (int k = 0; k < CH; ++k) {
        __builtin_nontemporal_store(V2, op + k * N2);   // out never re-read
        float z    = z2buf[k][tid];
        float vv   = V2 * V2;
        float coef = fmaf(-dt3, vv, one_dt);
        float Vn   = fmaf(V2, coef, FHN_DT * (z - S2));
        float Sn   = fmaf(FHN_DT * 0.08f, fmaf(-0.8f, S2, V2 + 0.7f), S2);
        V2 = Vn; S2 = Sn;
      }
    }
    __syncthreads();
  }
}

// ---------------------------------------------------------------------------
extern "C" void kernel_launch(void* const* d_in, const int* in_sizes, int n_in,
                              void* d_out, int out_size, void* d_ws, size_t ws_size,
                              hipStream_t stream) {
  const float* batch = (const float*)d_in[0];   // (64, 2048, 784)
  const float* W1    = (const float*)d_in[1];   // (100, 784)
  const float* W2    = (const float*)d_in[2];   // (10, 100)

  // workspace layout: z1 (52,428,800 B) | W1t packed (351,232 B)
  float* z1  = (float*)d_ws;
  float* W1t = (float*)((char*)d_ws + (size_t)(B_SZ * T_SZ) * N1 * sizeof(float));

  fhn_pack_w1   <<<dim3((N1P * K1 + 255) / 256), dim3(256), 0, stream>>>(W1, W1t);
  fhn_gemm1_wmma<<<dim3(512),  dim3(256), 0, stream>>>(batch, W1t, z1);
  fhn_scan_fused<<<dim3(B_SZ), dim3(128), 0, stream>>>(z1, W2, (float*)d_out);
}